// QSAR_43370579755168
// MI455X (gfx1250) — compile-verified
//
#include <hip/hip_runtime.h>
#include <hip/hip_bf16.h>
#include <math.h>

// ---------- problem constants (from reference setup_inputs) ----------
#define BB   128   // batch
#define NM   64    // molecule nodes
#define DM   6     // molecule degree
#define FM   37    // molecule atom features
#define FB   6     // bond features
#define NP   256   // protein nodes
#define DP   8     // protein degree
#define FPF  480   // protein atom features
#define HID  128

typedef __attribute__((ext_vector_type(2))) float v2f;
typedef __attribute__((ext_vector_type(8))) float v8f;

enum { ACT_NONE = 0, ACT_RELU = 1, ACT_TANH = 2, ACT_SIGMOID = 3 };

#define CT 4   // 16-col tiles per wave -> 16x64 output tile per wave

// =====================================================================
// Branch-free f32 WMMA GEMM.
//   A  : [M, lda] f32, rows mult of 16, pad cols (lda >= Kp) zero-filled
//   Bpk: packed weights, layout Bpk[(k>>2)*Np*4 + col*4 + (k&3)], zeros in pad
//   biasP: [Np] zero-padded bias
//   C  : [M, ldc] with ldc == Np (mult of 64)
// One wave -> 16x64 tile. Inner loop: 1 A b64 load + 4 B b64 loads + 4 WMMA.
// EXEC stays all-ones (no divergent control flow).
// =====================================================================
__global__ __launch_bounds__(32)
void wmma_gemm_f32(const float* __restrict__ A, int lda,
                   const float* __restrict__ Bpk, int Np,
                   const float* __restrict__ biasP,
                   float* __restrict__ C, int ldc,
                   int Kp, int act) {
  const int lane = threadIdx.x & 31;
  const int half = lane >> 4;     // K-pair select: 0 -> K{0,1}, 1 -> K{2,3}
  const int l15  = lane & 15;     // A row / B,D column within 16-tile
  const int row0 = blockIdx.x * 16;
  const int col0 = blockIdx.y * (16 * CT);

  const float* __restrict__ ap = A + (size_t)(row0 + l15) * lda + 2 * half;
  const float* __restrict__ bp = Bpk + (size_t)(col0 + l15) * 4 + 2 * half;
  const size_t bstep = (size_t)Np * 4;

  v8f acc[CT];
#pragma unroll
  for (int t = 0; t < CT; ++t) acc[t] = (v8f){0.f,0.f,0.f,0.f,0.f,0.f,0.f,0.f};

  for (int kb = 0; kb < Kp; kb += 4) {
    const v2f a = *reinterpret_cast<const v2f*>(ap);
    ap += 4;
#pragma unroll
    for (int t = 0; t < CT; ++t) {
      const v2f b = *reinterpret_cast<const v2f*>(bp + t * 64);  // 16 cols * 4
      acc[t] = __builtin_amdgcn_wmma_f32_16x16x4_f32(
          false, a, false, b, (short)0, acc[t], false, false);
    }
    bp += bstep;
  }

#pragma unroll
  for (int t = 0; t < CT; ++t) {
    const int col = col0 + t * 16 + l15;
    const float bv = biasP[col];
#pragma unroll
    for (int v = 0; v < 8; ++v) {
      const int r = row0 + v + 8 * half;      // D row: M = v + 8*half
      float x = acc[t][v] + bv;
      if      (act == ACT_RELU)    x = fmaxf(x, 0.f);
      else if (act == ACT_TANH)    x = tanhf(x);
      else if (act == ACT_SIGMOID) x = 1.f / (1.f + expf(-x));
      C[(size_t)r * ldc + col] = x;
    }
  }
}

// =====================================================================
// Pack a row-major weight [K,N] (+bias[N]) into WMMA-fragment order with
// zero padding: Bpk[(k>>2)*Np + n][k&3], biasP[Np]. Kp mult of 4, Np mult 64.
// =====================================================================
__global__ void pack_w(const float* __restrict__ W, const float* __restrict__ bias,
                       int K, int N, float* __restrict__ Bpk,
                       float* __restrict__ biasP, int Kp, int Np) {
  const int idx = blockIdx.x * blockDim.x + threadIdx.x;
  if (idx < Np) biasP[idx] = (idx < N) ? bias[idx] : 0.f;
  if (idx >= Kp * Np) return;
  const int k = idx / Np, n = idx % Np;
  const float v = (k < K && n < N) ? W[(size_t)k * N + n] : 0.f;
  Bpk[((size_t)(k >> 2) * Np + n) * 4 + (k & 3)] = v;
}

// =====================================================================
// Neighbor-sum gather (self + valid neighbors), float4 over F.
// One block per row. Writes degree mask (optional) and fills dst columns
// [F, dstStride) with `extra` (extraF values) then zeros (pad).
// =====================================================================
__global__ void neighbor_sum_f4(const float* __restrict__ src,
                                const int* __restrict__ edges,
                                float* __restrict__ dst,
                                int nodes, int D, int F, int srcStride, int dstStride,
                                float* __restrict__ mask,
                                const float* __restrict__ extra, int extraF) {
  const int row = blockIdx.x;           // b*nodes + n
  const int b   = row / nodes;
  const int* e  = edges + (size_t)row * D;

  int eid[8];
  int deg = 0;
#pragma unroll 8
  for (int d = 0; d < 8; ++d) {
    int v = (d < D) ? e[d] : -1;
    eid[d] = v;
    if (v >= 0) deg++;
  }

  const float* selfRow   = src + (size_t)row * srcStride;
  const float* batchBase = src + (size_t)b * nodes * srcStride;
  const int nch = F >> 2;

  for (int c = threadIdx.x; c < nch; c += blockDim.x) {
    float4 acc = *reinterpret_cast<const float4*>(selfRow + 4 * c);
    for (int d = 0; d < D; ++d) {
      const int v = eid[d];
      if (v >= 0) {
        float4 g = *reinterpret_cast<const float4*>(
            batchBase + (size_t)v * srcStride + 4 * c);
        acc.x += g.x; acc.y += g.y; acc.z += g.z; acc.w += g.w;
      }
    }
    *reinterpret_cast<float4*>(dst + (size_t)row * dstStride + 4 * c) = acc;
  }

  for (int t = threadIdx.x; t < dstStride - F; t += blockDim.x)
    dst[(size_t)row * dstStride + F + t] =
        (t < extraF) ? extra[(size_t)row * extraF + t] : 0.f;
  if (mask && threadIdx.x == 0)
    mask[row] = (deg > 0) ? 1.f : 0.f;
}

// =====================================================================
// Molecule prep: sb = sum_d m_bonds, x1 = [neighbor_sum(m_atoms)(37), sb(6), 0]
// stride 44, degree mask. One thread per (b, n).
// =====================================================================
__global__ void mol_prep(const float* __restrict__ m_atoms,
                         const float* __restrict__ m_bonds,
                         const int* __restrict__ m_edges,
                         float* __restrict__ x1, float* __restrict__ sb,
                         float* __restrict__ maskM) {
  const int row = blockIdx.x * blockDim.x + threadIdx.x;
  if (row >= BB * NM) return;
  const int b = row / NM;
  const int* e = m_edges + (size_t)row * DM;

  int eid[DM];
  int deg = 0;
#pragma unroll
  for (int d = 0; d < DM; ++d) { int v = e[d]; eid[d] = v; if (v >= 0) deg++; }
  maskM[row] = (deg > 0) ? 1.f : 0.f;

#pragma unroll
  for (int f = 0; f < FB; ++f) {
    float s = 0.f;
#pragma unroll
    for (int d = 0; d < DM; ++d)
      s += m_bonds[((size_t)row * DM + d) * FB + f];
    sb[(size_t)row * FB + f] = s;
    x1[(size_t)row * 44 + FM + f] = s;
  }
  x1[(size_t)row * 44 + 43] = 0.f;   // K padding

  const float* batchBase = m_atoms + (size_t)b * NM * FM;
  for (int f = 0; f < FM; ++f) {
    float s = m_atoms[(size_t)row * FM + f];
#pragma unroll
    for (int d = 0; d < DM; ++d) {
      int v = eid[d];
      if (v >= 0) s += batchBase[(size_t)v * FM + f];
    }
    x1[(size_t)row * 44 + f] = s;
  }
}

// x3 = [h2 (128), sb (6), 0, 0], stride 136. One block (32 threads) per row.
__global__ void concat_h_sb(const float* __restrict__ h,
                            const float* __restrict__ sb,
                            float* __restrict__ x3) {
  const int row = blockIdx.x;
  const int t = threadIdx.x;
  float4 v = *reinterpret_cast<const float4*>(h + (size_t)row * HID + 4 * t);
  *reinterpret_cast<float4*>(x3 + (size_t)row * 136 + 4 * t) = v;
  if (t < 8)
    x3[(size_t)row * 136 + HID + t] = (t < FB) ? sb[(size_t)row * FB + t] : 0.f;
}

// fp[b, outOff + j] = sum_n mask[b,n] * tmp[(b,n), j]   (j < 128)
__global__ void fp_reduce(const float* __restrict__ tmp,
                          const float* __restrict__ mask,
                          float* __restrict__ fp, int nodes, int outOff) {
  const int b = blockIdx.x;
  const int j = threadIdx.x;   // 128
  float s = 0.f;
  for (int n = 0; n < nodes; ++n) {
    const size_t r = (size_t)b * nodes + n;
    s += mask[r] * tmp[r * HID + j];
  }
  fp[(size_t)b * 256 + outOff + j] = s;
}

// 2-class softmax; logits stride 64
__global__ void softmax2(const float* __restrict__ logits, float* __restrict__ out) {
  const int i = blockIdx.x * blockDim.x + threadIdx.x;
  if (i >= BB) return;
  const float a = logits[(size_t)i * 64 + 0], b = logits[(size_t)i * 64 + 1];
  const float m = fmaxf(a, b);
  const float ea = expf(a - m), eb = expf(b - m);
  const float s = ea + eb;
  out[2 * i]     = ea / s;
  out[2 * i + 1] = eb / s;
}

// =====================================================================
extern "C" void kernel_launch(void* const* d_in, const int* in_sizes, int n_in,
                              void* d_out, int out_size, void* d_ws, size_t ws_size,
                              hipStream_t stream) {
  const float* m_atoms = (const float*)d_in[0];
  const float* m_bonds = (const float*)d_in[1];
  const int*   m_edges = (const int*)  d_in[2];
  const float* p_atoms = (const float*)d_in[3];
  const int*   p_edges = (const int*)  d_in[4];
  const float* Wm1  = (const float*)d_in[5];  const float* bm1  = (const float*)d_in[6];
  const float* Wm2  = (const float*)d_in[7];  const float* bm2  = (const float*)d_in[8];
  const float* Wgop = (const float*)d_in[9];  const float* bgop = (const float*)d_in[10];
  const float* Wp1  = (const float*)d_in[11]; const float* bp1  = (const float*)d_in[12];
  const float* Wp2  = (const float*)d_in[13]; const float* bp2  = (const float*)d_in[14];
  const float* Wgp  = (const float*)d_in[15]; const float* bgp  = (const float*)d_in[16];
  const float* Wfc1 = (const float*)d_in[17]; const float* bfc1 = (const float*)d_in[18];
  const float* Wfc2 = (const float*)d_in[19]; const float* bfc2 = (const float*)d_in[20];
  float* out = (float*)d_out;
  float* ws = (float*)d_ws;
  (void)ws_size; (void)in_sizes; (void)n_in; (void)out_size;

  const int MP = BB * NP;   // 32768 protein rows
  const int MM = BB * NM;   // 8192 molecule rows

  // ---- workspace layout (floats, 16-float aligned, with buffer reuse) ----
  const size_t oNsP0 = 0;                          // 32768*480  (region A)
  const size_t oG1   = (size_t)MP * FPF;           // 32768*256  (region B)
  const size_t oNsP1 = 0;                          // reuse A: 32768*200
  const size_t oG2   = (size_t)MP * 200;           // in A:    32768*128
  const size_t oTmpP = oG1;                        // reuse B: 32768*128
  size_t o = oG1 + (size_t)MP * 256;               // region C
  auto alloc = [&o](size_t n) { size_t r = o; o += (n + 15) & ~(size_t)15; return r; };
  const size_t oX1    = alloc((size_t)MM * 44);
  const size_t oH1    = alloc((size_t)MM * HID);
  const size_t oX2    = alloc((size_t)MM * 136);
  const size_t oH2    = alloc((size_t)MM * HID);
  const size_t oX3    = alloc((size_t)MM * 136);
  const size_t oTmpM  = alloc((size_t)MM * HID);
  const size_t oSb    = alloc((size_t)MM * FB);
  const size_t oMaskM = alloc(MM);
  const size_t oMaskP = alloc(MP);
  const size_t oFp    = alloc((size_t)BB * 256);
  const size_t oInter = alloc((size_t)BB * 128);
  const size_t oLogit = alloc((size_t)BB * 64);
  // packed weights + padded biases
  struct PW { size_t w, b; int K, N, Kp, Np; };
  auto mk = [&](int K, int N, int Kp, int Np) {
    PW p; p.K = K; p.N = N; p.Kp = Kp; p.Np = Np;
    p.w = alloc((size_t)Kp * Np); p.b = alloc(Np); return p;
  };
  const PW pm1 = mk(FM + FB, HID, 44, 128);      // Wm1  43 x128
  const PW pm2 = mk(HID + FB, HID, 136, 128);    // Wm2  134x128
  const PW pmg = mk(HID + FB, HID, 136, 128);    // Wgop 134x128
  const PW pp1 = mk(FPF, 200, 480, 256);         // Wp1  480x200
  const PW pp2 = mk(200, 100, 200, 128);         // Wp2  200x100
  const PW ppg = mk(100, HID, 100, 128);         // Wgp  100x128
  const PW pf1 = mk(256, 100, 256, 128);         // Wfc1 256x100
  const PW pf2 = mk(100, 2, 100, 64);            // Wfc2 100x2

  auto pack = [&](const PW& p, const float* W, const float* bias) {
    const int tot = p.Kp * p.Np;
    pack_w<<<(tot + 255) / 256, 256, 0, stream>>>(W, bias, p.K, p.N,
                                                  ws + p.w, ws + p.b, p.Kp, p.Np);
  };
  pack(pm1, Wm1, bm1);  pack(pm2, Wm2, bm2);  pack(pmg, Wgop, bgop);
  pack(pp1, Wp1, bp1);  pack(pp2, Wp2, bp2);  pack(ppg, Wgp, bgp);
  pack(pf1, Wfc1, bfc1); pack(pf2, Wfc2, bfc2);

  auto gemm = [&](size_t oA, int lda, const PW& p, size_t oC, int M, int act) {
    wmma_gemm_f32<<<dim3(M / 16, p.Np / 64), 32, 0, stream>>>(
        ws + oA, lda, ws + p.w, p.Np, ws + p.b, ws + oC, p.Np, p.Kp, act);
  };

  // ================= protein branch =================
  neighbor_sum_f4<<<MP, 128, 0, stream>>>(p_atoms, p_edges, ws + oNsP0,
                                          NP, DP, FPF, FPF, FPF,
                                          ws + oMaskP, nullptr, 0);
  gemm(oNsP0, FPF, pp1, oG1, MP, ACT_RELU);                 // g1 [32768,256] (200 live)
  neighbor_sum_f4<<<MP, 64, 0, stream>>>(ws + oG1, p_edges, ws + oNsP1,
                                         NP, DP, 200, 256, 200,
                                         nullptr, nullptr, 0);
  gemm(oNsP1, 200, pp2, oG2, MP, ACT_RELU);                 // g2 [32768,128] (100 live)
  gemm(oG2, 128, ppg, oTmpP, MP, ACT_TANH);                 // tmpP [32768,128]
  fp_reduce<<<BB, HID, 0, stream>>>(ws + oTmpP, ws + oMaskP, ws + oFp, NP, 128);

  // ================= molecule branch =================
  mol_prep<<<(MM + 255) / 256, 256, 0, stream>>>(m_atoms, m_bonds, m_edges,
                                                 ws + oX1, ws + oSb, ws + oMaskM);
  gemm(oX1, 44, pm1, oH1, MM, ACT_RELU);                    // h1 [8192,128]
  neighbor_sum_f4<<<MM, 32, 0, stream>>>(ws + oH1, m_edges, ws + oX2,
                                         NM, DM, HID, HID, 136,
                                         nullptr, ws + oSb, FB);
  gemm(oX2, 136, pm2, oH2, MM, ACT_RELU);                   // h2 [8192,128]
  concat_h_sb<<<MM, 32, 0, stream>>>(ws + oH2, ws + oSb, ws + oX3);
  gemm(oX3, 136, pmg, oTmpM, MM, ACT_TANH);                 // tmpM [8192,128]
  fp_reduce<<<BB, HID, 0, stream>>>(ws + oTmpM, ws + oMaskM, ws + oFp, NM, 0);

  // ================= classifier =================
  gemm(oFp, 256, pf1, oInter, BB, ACT_SIGMOID);             // inter [128,128] (100 live)
  gemm(oInter, 128, pf2, oLogit, BB, ACT_NONE);             // logits [128,64] (2 live)
  softmax2<<<1, BB, 0, stream>>>(ws + oLogit, out);
}